// ELECT_Mnist_64080912056481
// MI455X (gfx1250) — compile-verified
//
#include <hip/hip_runtime.h>
#include <hip/hip_bf16.h>
#include <math.h>

// ---------------- problem constants ----------------
#define NGRAPH   256
#define NPNODE   128
#define NNODE    32768
#define DDIM     512
#define EHALFTOT 262144   // G * E_HALF
#define ETOT     524288   // total directed edges
#define H1DIM    64
#define LEAKY_SLOPE 0.01f
#define INV_SQRT_N  0.08838834764831845f   // 1/sqrt(128), every graph has 128 nodes
#define BN_EPS_C    1e-5f

typedef __attribute__((ext_vector_type(2))) float v2f;
typedef __attribute__((ext_vector_type(8))) float v8f;
typedef __attribute__((ext_vector_type(4))) int   v4i;

#if __has_builtin(__builtin_amdgcn_global_load_async_to_lds_b128)
#define HAVE_ASYNC_LDS 1
#else
#define HAVE_ASYNC_LDS 0
#endif

#if HAVE_ASYNC_LDS
typedef __attribute__((address_space(1))) v4i gv4i;   // global 128-bit block
typedef __attribute__((address_space(3))) v4i lv4i;   // LDS 128-bit block
__device__ __forceinline__ void wait_asynccnt0()
{
#if __has_builtin(__builtin_amdgcn_s_wait_asynccnt)
    __builtin_amdgcn_s_wait_asynccnt(0);
#else
    asm volatile("s_wait_asynccnt 0x0" ::: "memory");
#endif
}
#endif

// =====================================================================
// GEMM with fp32 WMMA (V_WMMA_F32_16X16X4_F32): Out = act(A @ W + bias)
// grid: (M/128, N/64); block 256 = 8 waves; wave w owns rows [w*16,w*16+16)
// of the 128-row tile and all 64 columns (4 accumulators).
// A tile staged 128x32 row-major (async-to-LDS when available);
// W tile staged TRANSPOSED Wt[n][k] (stride 34) so both A and B fragments
// are contiguous (k,k+1) pairs -> single ds_load_b64 per fragment.
// =====================================================================
#define WT_STRIDE 34

template<bool LEAKY>
__global__ void k_gemm_bias_act(const float* __restrict__ A, const float* __restrict__ W,
                                const float* __restrict__ bias, float* __restrict__ Out,
                                int K, int N)
{
    __shared__ float As[128 * 32];            // 16 KB
    __shared__ float Wt[64 * WT_STRIDE];      // 8.5 KB
    const int tid  = threadIdx.x;
    const int lane = tid & 31;
    const int wv   = tid >> 5;
    const int m0   = blockIdx.x * 128;
    const int n0   = blockIdx.y * 64;

    v8f acc[4] = {};

    for (int kk = 0; kk < K; kk += 32) {
        // ---- stage A tile (128 x 32) ----
#if HAVE_ASYNC_LDS
        {
#pragma unroll
            for (int it = 0; it < 4; ++it) {
                int bi   = tid + 256 * it;        // 1024 x b128 blocks
                int r    = bi >> 3;               // 8 blocks per 32-float row
                int part = bi & 7;
                __builtin_amdgcn_global_load_async_to_lds_b128(
                    (gv4i*)(A + (size_t)(m0 + r) * K + kk + part * 4),
                    (lv4i*)&As[bi * 4], 0, 0);
            }
        }
#else
        for (int i = tid; i < 128 * 32; i += 256) {
            int r = i >> 5, c = i & 31;
            As[r * 32 + c] = A[(size_t)(m0 + r) * K + kk + c];
        }
#endif
        // ---- stage W tile transposed: Wt[c][r] = W[kk+r][n0+c] ----
        for (int i = tid; i < 32 * 64; i += 256) {
            int r = i >> 6, c = i & 63;
            Wt[c * WT_STRIDE + r] = W[(size_t)(kk + r) * N + n0 + c];
        }
#if HAVE_ASYNC_LDS
        wait_asynccnt0();
#endif
        __syncthreads();

#pragma unroll
        for (int k4 = 0; k4 < 8; ++k4) {
            // lanes 0-15 hold K = kb,kb+1 ; lanes 16-31 hold K = kb+2,kb+3
            int kb = k4 * 4 + ((lane >> 4) << 1);
            int mr = wv * 16 + (lane & 15);
            v2f a = *(const v2f*)&As[mr * 32 + kb];
            int nc = lane & 15;
#pragma unroll
            for (int t = 0; t < 4; ++t) {
                v2f b = *(const v2f*)&Wt[(nc + t * 16) * WT_STRIDE + kb];
                acc[t] = __builtin_amdgcn_wmma_f32_16x16x4_f32(
                    false, a, false, b, (short)0, acc[t], false, false);
            }
        }
        __syncthreads();
    }

    // C/D layout: VGPR r -> lanes 0-15: M=r, lanes 16-31: M=r+8; N = lane&15
    int rowBase = m0 + wv * 16 + ((lane >> 4) << 3);
    int colBase = n0 + (lane & 15);
#pragma unroll
    for (int t = 0; t < 4; ++t) {
        int col = colBase + t * 16;
        float bv = bias[col];
#pragma unroll
        for (int r = 0; r < 8; ++r) {
            float v = acc[t][r] + bv;
            v = LEAKY ? (v > 0.f ? v : LEAKY_SLOPE * v) : fmaxf(v, 0.f);
            Out[(size_t)(rowBase + r) * N + col] = v;
        }
    }
}

// ---------------- conv1 helpers (features are scalar) ----------------
__global__ void k_edge_agg_scalar(const float* __restrict__ x, const int* __restrict__ row,
                                  const int* __restrict__ col, float* __restrict__ agg)
{
    int e = blockIdx.x * 256 + threadIdx.x;
    if (e < ETOT) atomicAdd(&agg[col[e]], x[row[e]]);
}

__global__ void k_conv1_expand(const float* __restrict__ x, const float* __restrict__ agg,
                               const float* __restrict__ eps, const float* __restrict__ w1,
                               const float* __restrict__ b1, float* __restrict__ Out)
{
    int i = blockIdx.x * 256 + threadIdx.x;     // i < N*D
    int n = i >> 9, d = i & 511;
    float z = (1.f + eps[0]) * x[n] + agg[n];
    float v = z * w1[d] + b1[d];
    Out[i] = fmaxf(v, 0.f);
}

// ---------------- per-graph LDS message passing ----------------
// grid (graph, half); block 256; LDS: h-tile 128x256 + agg-tile 128x256 + 2048 packed idx
__global__ void k_gather_agg(const float* __restrict__ H, const int* __restrict__ row,
                             const int* __restrict__ col, const float* __restrict__ eps,
                             float* __restrict__ Out)
{
    extern __shared__ float smem[];
    float* hs  = smem;                 // 32768 floats
    float* as  = smem + 32768;         // 32768 floats
    int*   idx = (int*)(smem + 65536); // 2048 ints
    const int tid  = threadIdx.x;
    const int g    = blockIdx.x;
    const int half = blockIdx.y;
    const int cbase = half * 256;

    for (int i = tid; i < 32768; i += 256) {
        int node = i >> 8, c = i & 255;
        hs[i] = H[(size_t)(g * 128 + node) * DDIM + cbase + c];
        as[i] = 0.f;
    }
    for (int j = tid; j < 2048; j += 256) {
        int eg = (j < 1024) ? (g * 1024 + j) : (EHALFTOT + g * 1024 + (j - 1024));
        int lr = row[eg] - g * 128;
        int lc = col[eg] - g * 128;
        idx[j] = lr | (lc << 16);
    }
    __syncthreads();
    const int c = tid;   // each thread owns one channel -> conflict-free accumulation
    for (int e = 0; e < 2048; ++e) {
        int pk = idx[e];
        int lr = pk & 0xffff;
        int lc = pk >> 16;
        as[lc * 256 + c] += hs[lr * 256 + c];
    }
    __syncthreads();
    float ep = 1.f + eps[0];
    for (int i = tid; i < 32768; i += 256) {
        int node = i >> 8, c2 = i & 255;
        Out[(size_t)(g * 128 + node) * DDIM + cbase + c2] = ep * hs[i] + as[i];
    }
}

// ---------------- batchnorm stats / apply ----------------
__global__ void k_colstats(const float* __restrict__ X, float* gsum, float* gsq)
{
    int ch = blockIdx.x * 256 + threadIdx.x;
    int r0 = blockIdx.y * 512;
    float s = 0.f, q = 0.f;
    for (int r = 0; r < 512; ++r) {
        float v = X[(size_t)(r0 + r) * DDIM + ch];
        s += v; q += v * v;
    }
    atomicAdd(&gsum[ch], s);
    atomicAdd(&gsq[ch], q);
}

// MODE 0: Out = BN(X)
// MODE 1: Out = leaky(BN(X)) * scale            (+collect stats)
// MODE 2: Out = (R + leaky(BN(X))) * scale      (+collect stats)
template<int MODE>
__global__ void k_bn_apply(const float* __restrict__ X, const float* __restrict__ R,
                           float* __restrict__ Out,
                           const float* __restrict__ sumin, const float* __restrict__ sqin,
                           const float* __restrict__ g, const float* __restrict__ b,
                           float* sumout, float* sqout, float scale)
{
    int ch = blockIdx.x * 256 + threadIdx.x;
    int r0 = blockIdx.y * 512;
    const float invN = 1.f / 32768.f;
    float m   = sumin[ch] * invN;
    float var = sqin[ch] * invN - m * m;
    float rs  = rsqrtf(var + BN_EPS_C);
    float gg = g[ch], bb = b[ch];
    float s = 0.f, q = 0.f;
    for (int r = 0; r < 512; ++r) {
        size_t i = (size_t)(r0 + r) * DDIM + ch;
        float y = (X[i] - m) * rs * gg + bb;
        if (MODE == 1) { y = (y > 0.f ? y : LEAKY_SLOPE * y) * scale; s += y; q += y * y; }
        if (MODE == 2) { y = (R[i] + (y > 0.f ? y : LEAKY_SLOPE * y)) * scale; s += y; q += y * y; }
        Out[i] = y;
    }
    if (MODE != 0) { atomicAdd(&sumout[ch], s); atomicAdd(&sqout[ch], q); }
}

// ---------------- head ----------------
__global__ void k_lin2(const float* __restrict__ E, const float* __restrict__ w,
                       const float* __restrict__ b, float* __restrict__ h2)
{
    int n = blockIdx.x * 256 + threadIdx.x;
    float acc = b[0];
    for (int j = 0; j < 64; ++j) acc += E[(size_t)n * 64 + j] * w[j];
    h2[n] = acc > 0.f ? acc : LEAKY_SLOPE * acc;
}

__global__ void k_minmax_p(const float* __restrict__ h2, float* __restrict__ p_out)
{
    __shared__ float smx[128], smn[128];
    int g = blockIdx.x, tid = threadIdx.x;
    float v = h2[g * 128 + tid];
    smx[tid] = v; smn[tid] = v;
    __syncthreads();
    for (int s = 64; s > 0; s >>= 1) {
        if (tid < s) { smx[tid] = fmaxf(smx[tid], smx[tid + s]); smn[tid] = fminf(smn[tid], smn[tid + s]); }
        __syncthreads();
    }
    float mx = smx[0], mn = smn[0];
    p_out[g * 128 + tid] = (v - mn) / (mx + 1e-6f - mn);
}

__global__ void k_edge_reduce(const float* __restrict__ p, const int* __restrict__ row,
                              const int* __restrict__ col, const float* __restrict__ attr,
                              float* __restrict__ pairsum, float* __restrict__ gammaG)
{
    __shared__ float s1[256], s2[256];
    int g = blockIdx.x, tid = threadIdx.x;
    float pw = 0.f, gm = 0.f;
    for (int j = tid; j < 2048; j += 256) {
        int eg = (j < 1024) ? (g * 1024 + j) : (EHALFTOT + g * 1024 + (j - 1024));
        float d = p[row[eg]] - p[col[eg]];
        float w = attr[eg];
        pw += d * d * w; gm += w;
    }
    s1[tid] = pw; s2[tid] = gm;
    __syncthreads();
    for (int s = 128; s > 0; s >>= 1) {
        if (tid < s) { s1[tid] += s1[tid + s]; s2[tid] += s2[tid + s]; }
        __syncthreads();
    }
    if (tid == 0) { pairsum[g] = 0.5f * s1[0]; gammaG[g] = 0.5f * s2[0]; }
}

// ---------------- eigen machinery (Householder + Sturm bisection) ----------------
__device__ __forceinline__ float blockReduceSum128(float v, float* red, int tid)
{
    red[tid] = v;
    __syncthreads();
    for (int s = 64; s > 0; s >>= 1) {
        if (tid < s) red[tid] += red[tid + s];
        __syncthreads();
    }
    float r = red[0];
    __syncthreads();
    return r;
}

__device__ void tridiag128(float* mat, float* dd, float* ee,
                           float* vv, float* pp, float* qq, float* red, int tid)
{
    const int n = 128;
    for (int k = 0; k < n - 2; ++k) {
        float x = (tid > k) ? mat[tid * n + k] : 0.f;
        float sigma = blockReduceSum128(x * x, red, tid);
        float alpha = mat[(k + 1) * n + k];
        if (sigma < 1e-24f) {
            if (tid == 0) ee[k] = alpha;
            __syncthreads();
            continue;
        }
        float s = sqrtf(sigma);
        if (alpha > 0.f) s = -s;
        float h = sigma - alpha * s;
        if (tid == 0) ee[k] = s;
        vv[tid] = (tid <= k) ? 0.f : ((tid == k + 1) ? (alpha - s) : mat[tid * n + k]);
        __syncthreads();
        float acc = 0.f;
        if (tid > k) {
            for (int j = k + 1; j < n; ++j) acc += mat[tid * n + j] * vv[j];
            acc /= h;
        }
        pp[tid] = acc;
        __syncthreads();
        float Kc = blockReduceSum128(vv[tid] * pp[tid], red, tid) / (2.f * h);
        qq[tid] = pp[tid] - Kc * vv[tid];
        __syncthreads();
        if (tid > k) {
            float vt = vv[tid], qt = qq[tid];
            for (int j = k + 1; j < n; ++j)
                mat[tid * n + j] -= vt * qq[j] + qt * vv[j];
        }
        __syncthreads();
    }
    dd[tid] = mat[tid * n + tid];
    if (tid == 0) { ee[n - 2] = mat[(n - 1) * n + (n - 2)]; ee[n - 1] = 0.f; }
    __syncthreads();
}

__device__ int sturmCount(const float* dd, const float* ee, float x)
{
    int cnt = 0;
    float q = 1.f;
    for (int i = 0; i < 128; ++i) {
        float e2 = (i == 0) ? 0.f : ee[i - 1] * ee[i - 1];
        q = dd[i] - x - e2 / q;
        if (q < 0.f) cnt++;
        if (fabsf(q) < 1e-20f) q = (q < 0.f) ? -1e-20f : 1e-20f;
    }
    return cnt;
}

__device__ float bisectEig(const float* dd, const float* ee, int idx, float lo, float hi)
{
    for (int it = 0; it < 60; ++it) {
        float mid = 0.5f * (lo + hi);
        if (sturmCount(dd, ee, mid) > idx) hi = mid; else lo = mid;
    }
    return 0.5f * (lo + hi);
}

__device__ void gershgorin(const float* dd, const float* ee, float* lo, float* hi)
{
    float l = 1e30f, h = -1e30f;
    for (int i = 0; i < 128; ++i) {
        float r = ((i > 0) ? fabsf(ee[i - 1]) : 0.f) + fabsf(ee[i]);
        l = fminf(l, dd[i] - r);
        h = fmaxf(h, dd[i] + r);
    }
    *lo = l; *hi = h;
}

// lap0 eigensolve -> aaa[g] = -log(1e-4)/ev2
__global__ void k_eig0(const int* __restrict__ row, const int* __restrict__ col,
                       float* __restrict__ aaa)
{
    extern __shared__ float smem[];
    float* mat = smem;
    float* dd = smem + 16384; float* ee = dd + 128; float* vv = ee + 128;
    float* pp = vv + 128;     float* qq = pp + 128; float* red = qq + 128;
    int g = blockIdx.x, tid = threadIdx.x;
    for (int i = tid; i < 16384; i += 128) mat[i] = 0.f;
    __syncthreads();
    for (int j = tid; j < 2048; j += 128) {
        int eg = (j < 1024) ? (g * 1024 + j) : (EHALFTOT + g * 1024 + (j - 1024));
        int lr = row[eg] - g * 128;
        int lc = col[eg] - g * 128;
        atomicAdd(&mat[lr * 128 + lc], 1.0f);
    }
    __syncthreads();
    float deg = 0.f;
    for (int j = 0; j < 128; ++j) deg += mat[tid * 128 + j];
    for (int j = 0; j < 128; ++j) {
        float v = mat[tid * 128 + j];
        mat[tid * 128 + j] = (j == tid ? deg : 0.f) - v;
    }
    __syncthreads();
    tridiag128(mat, dd, ee, vv, pp, qq, red, tid);
    if (tid == 0) {
        float lo, hi; gershgorin(dd, ee, &lo, &hi);
        float ev2 = bisectEig(dd, ee, 2, lo, hi);
        aaa[g] = 9.210340371976184f / ev2;   // log(1e-4)/(-ev2)
    }
}

// modified-lap eigensolve -> final loss terms into d_out
__global__ void k_eig1(const int* __restrict__ row, const int* __restrict__ col,
                       const float* __restrict__ p, const float* __restrict__ aaa,
                       const float* __restrict__ pairsum, const float* __restrict__ gammaG,
                       float* __restrict__ out)
{
    extern __shared__ float smem[];
    float* mat = smem;
    float* dd = smem + 16384; float* ee = dd + 128; float* vv = ee + 128;
    float* pp = vv + 128;     float* qq = pp + 128; float* red = qq + 128;
    int g = blockIdx.x, tid = threadIdx.x;
    for (int i = tid; i < 16384; i += 128) mat[i] = 0.f;
    __syncthreads();
    for (int j = tid; j < 2048; j += 128) {
        int eg = (j < 1024) ? (g * 1024 + j) : (EHALFTOT + g * 1024 + (j - 1024));
        int lr = row[eg] - g * 128;
        int lc = col[eg] - g * 128;
        float t = 1.f - p[row[eg]] - p[col[eg]];
        atomicAdd(&mat[lr * 128 + lc], t * t);
    }
    __syncthreads();
    float deg = 0.f;
    for (int j = 0; j < 128; ++j) deg += mat[tid * 128 + j];
    for (int j = 0; j < 128; ++j) {
        float v = mat[tid * 128 + j];
        mat[tid * 128 + j] = (j == tid ? deg : 0.f) - v;
    }
    __syncthreads();
    tridiag128(mat, dd, ee, vv, pp, qq, red, tid);
    if (tid == 0) { gershgorin(dd, ee, &red[0], &red[1]); }
    __syncthreads();
    if (tid < 2) red[4 + tid] = bisectEig(dd, ee, 1 + tid, red[0], red[1]);
    __syncthreads();
    if (tid == 0) {
        float lam3 = red[5] - red[4];
        float fr = gammaG[g] - pairsum[g];
        float gr = expf(-aaa[g] * lam3);
        out[32768 + g] = (120.f / 36.f) * fr + gammaG[g] * gr;   // loss
        out[33025 + g] = fr;                                      // f_relax
        out[33281 + g] = gr;                                      // g_relax
    }
}

__global__ void k_mean(float* __restrict__ out)
{
    __shared__ float s[256];
    int tid = threadIdx.x;
    s[tid] = out[32768 + tid];
    __syncthreads();
    for (int st = 128; st > 0; st >>= 1) {
        if (tid < st) s[tid] += s[tid + st];
        __syncthreads();
    }
    if (tid == 0) out[33024] = s[0] * (1.f / 256.f);
}

// =====================================================================
extern "C" void kernel_launch(void* const* d_in, const int* in_sizes, int n_in,
                              void* d_out, int out_size, void* d_ws, size_t ws_size,
                              hipStream_t stream)
{
    (void)in_sizes; (void)n_in; (void)out_size; (void)ws_size;

    // pytree-flatten order (dicts alphabetical): x, edge_attr, params..., edge_index, batch, ebi
    const float* x         = (const float*)d_in[0];
    const float* edge_attr = (const float*)d_in[1];
    const float* bn1_b = (const float*)d_in[2];
    const float* bn1_g = (const float*)d_in[3];
    const float* bns_b[2] = {(const float*)d_in[4], (const float*)d_in[6]};
    const float* bns_g[2] = {(const float*)d_in[5], (const float*)d_in[7]};
    const float* c1_bn_b = (const float*)d_in[8];
    const float* c1_bn_g = (const float*)d_in[9];
    const float* c1_eps  = (const float*)d_in[10];
    const float* c1_l1_b = (const float*)d_in[11];
    const float* c1_l1_w = (const float*)d_in[12];
    const float* c1_l2_b = (const float*)d_in[13];
    const float* c1_l2_w = (const float*)d_in[14];
    const float *cv_bn_b[2], *cv_bn_g[2], *cv_eps[2], *cv_l1_b[2], *cv_l1_w[2], *cv_l2_b[2], *cv_l2_w[2];
    for (int i = 0; i < 2; ++i) {
        int base = 15 + 7 * i;
        cv_bn_b[i] = (const float*)d_in[base + 0];
        cv_bn_g[i] = (const float*)d_in[base + 1];
        cv_eps[i]  = (const float*)d_in[base + 2];
        cv_l1_b[i] = (const float*)d_in[base + 3];
        cv_l1_w[i] = (const float*)d_in[base + 4];
        cv_l2_b[i] = (const float*)d_in[base + 5];
        cv_l2_w[i] = (const float*)d_in[base + 6];
    }
    const float* lin1_b = (const float*)d_in[29];
    const float* lin1_w = (const float*)d_in[30];
    const float* lin2_b = (const float*)d_in[31];
    const float* lin2_w = (const float*)d_in[32];
    const int* ei   = (const int*)d_in[33];
    const int* erow = ei;
    const int* ecol = ei + ETOT;

    float* ws = (float*)d_ws;
    float* A = ws;                       // N x D
    float* B = ws + 16777216;            // N x D
    float* C = ws + 33554432;            // N x D
    float* agg0    = ws + 50331648;      // N
    float* h2      = agg0 + 32768;       // N
    float* sum0    = h2 + 32768;         // 512
    float* sq0     = sum0 + 512;         // 512
    float* sum1    = sq0 + 512;          // 512
    float* sq1     = sum1 + 512;         // 512
    float* pairsum = sq1 + 512;          // 256
    float* gammaG  = pairsum + 256;      // 256
    float* aaa     = gammaG + 256;       // 256
    float* outF = (float*)d_out;

    // allow large dynamic LDS (gather: 256KB+idx, eig: 67KB). Idempotent, capture-safe.
    (void)hipFuncSetAttribute((const void*)k_gather_agg, hipFuncAttributeMaxDynamicSharedMemorySize, 270336);
    (void)hipFuncSetAttribute((const void*)k_eig0,       hipFuncAttributeMaxDynamicSharedMemorySize, 69632);
    (void)hipFuncSetAttribute((const void*)k_eig1,       hipFuncAttributeMaxDynamicSharedMemorySize, 69632);

    // ---- conv1 ----
    (void)hipMemsetAsync(agg0, 0, 32768 * sizeof(float), stream);
    k_edge_agg_scalar<<<2048, 256, 0, stream>>>(x, erow, ecol, agg0);
    k_conv1_expand<<<65536, 256, 0, stream>>>(x, agg0, c1_eps, c1_l1_w, c1_l1_b, B);
    k_gemm_bias_act<false><<<dim3(256, 8), 256, 0, stream>>>(B, c1_l2_w, c1_l2_b, C, 512, 512);
    (void)hipMemsetAsync(sum0, 0, 1024 * sizeof(float), stream);
    k_colstats<<<dim3(2, 64), 256, 0, stream>>>(C, sum0, sq0);
    (void)hipMemsetAsync(sum1, 0, 1024 * sizeof(float), stream);
    k_bn_apply<1><<<dim3(2, 64), 256, 0, stream>>>(C, nullptr, B, sum0, sq0,
                                                   c1_bn_g, c1_bn_b, sum1, sq1, INV_SQRT_N);
    k_bn_apply<0><<<dim3(2, 64), 256, 0, stream>>>(B, nullptr, A, sum1, sq1,
                                                   bn1_g, bn1_b, nullptr, nullptr, 1.f);

    // ---- conv layers 2,3 ----
    for (int i = 0; i < 2; ++i) {
        k_gather_agg<<<dim3(256, 2), 256, 270336, stream>>>(A, erow, ecol, cv_eps[i], B);
        k_gemm_bias_act<false><<<dim3(256, 8), 256, 0, stream>>>(B, cv_l1_w[i], cv_l1_b[i], C, 512, 512);
        k_gemm_bias_act<false><<<dim3(256, 8), 256, 0, stream>>>(C, cv_l2_w[i], cv_l2_b[i], B, 512, 512);
        (void)hipMemsetAsync(sum0, 0, 1024 * sizeof(float), stream);
        k_colstats<<<dim3(2, 64), 256, 0, stream>>>(B, sum0, sq0);
        (void)hipMemsetAsync(sum1, 0, 1024 * sizeof(float), stream);
        k_bn_apply<2><<<dim3(2, 64), 256, 0, stream>>>(B, A, C, sum0, sq0,
                                                       cv_bn_g[i], cv_bn_b[i], sum1, sq1, INV_SQRT_N);
        k_bn_apply<0><<<dim3(2, 64), 256, 0, stream>>>(C, nullptr, A, sum1, sq1,
                                                       bns_g[i], bns_b[i], nullptr, nullptr, 1.f);
    }

    // ---- head ----
    k_gemm_bias_act<true><<<dim3(256, 1), 256, 0, stream>>>(A, lin1_w, lin1_b, C, 512, 64);
    k_lin2<<<128, 256, 0, stream>>>(C, lin2_w, lin2_b, h2);
    k_minmax_p<<<256, 128, 0, stream>>>(h2, outF);
    k_edge_reduce<<<256, 256, 0, stream>>>(outF, erow, ecol, edge_attr, pairsum, gammaG);
    k_eig0<<<256, 128, 69632, stream>>>(erow, ecol, aaa);
    k_eig1<<<256, 128, 69632, stream>>>(erow, ecol, outF, aaa, pairsum, gammaG, outF);
    k_mean<<<1, 256, 0, stream>>>(outF);
}